// FBP_21320217658069
// MI455X (gfx1250) — compile-verified
//
#include <hip/hip_runtime.h>
#include <hip/hip_bf16.h>
#include <math.h>

typedef __attribute__((ext_vector_type(2))) float v2f;
typedef __attribute__((ext_vector_type(8))) float v8f;

#define BB 2
#define VV 180
#define RR 32
#define CC 512
#define MM (BB*VV*RR)   // 11520 GEMM rows

// ---------------------------------------------------------------------------
// Kernel 1: ramp-filter impulse response h[n] = (1/N) * sum_f ramp[f]*cos(2pi f n/N)
// ramp[f] = 2*min(f, N-f)/N  (matches 2*|fftfreq(N)|). Exact mod-N phase reduction.
// ---------------------------------------------------------------------------
__global__ void fbp_ramp_h(float* __restrict__ h) {
    int n = blockIdx.x * blockDim.x + threadIdx.x;
    if (n >= CC) return;
    const float twopi_over_N = 6.28318530717958647692f / (float)CC;
    float acc = 0.0f;
    for (int f = 0; f < CC; ++f) {
        float ramp = 2.0f * fminf((float)f, (float)(CC - f)) * (1.0f / (float)CC);
        int ph = (f * n) & (CC - 1);            // (f*n) mod 512, exact in int
        acc += ramp * cosf(twopi_over_N * (float)ph);
    }
    h[n] = acc * (1.0f / (float)CC);
}

// ---------------------------------------------------------------------------
// Kernel 2: circulant matrix W[k][n] = h[(n - k) mod 512]
// ---------------------------------------------------------------------------
__global__ void fbp_circulant(const float* __restrict__ h, float* __restrict__ W) {
    int idx = blockIdx.x * blockDim.x + threadIdx.x;     // 512*512 total
    int k = idx >> 9;
    int n = idx & (CC - 1);
    W[idx] = h[(n - k) & (CC - 1)];
}

// ---------------------------------------------------------------------------
// Kernel 3: filtering GEMM  filt(11520x512) = sino(11520x512) @ W(512x512)
// One wave32 per 16x16 tile using V_WMMA_F32_16X16X4_F32, K-loop of 128 steps.
// A 16x4 layout: lanes 0-15 -> K={k0,k0+1}; lanes 16-31 -> K={k0+2,k0+3}.
// B 4x16 layout: v0: lanes0-15 row k0, lanes16-31 row k0+2; v1: rows k0+1/k0+3.
// D 16x16: VGPR j -> M = j (lanes<16) or 8+j (lanes>=16), N = lane&15.
// ---------------------------------------------------------------------------
__global__ void fbp_filter_gemm(const float* __restrict__ A,
                                const float* __restrict__ W,
                                float* __restrict__ out) {
    int wave = (blockIdx.x * blockDim.x + threadIdx.x) >> 5;  // 23040 waves
    int lane = threadIdx.x & 31;
    int mt = wave >> 5;          // 720 M-tiles
    int nt = wave & 31;          // 32  N-tiles
    int m0 = mt << 4;
    int n0 = nt << 4;
    int lm = lane & 15;
    int hi = (lane >> 4) & 1;    // 0: lanes 0-15, 1: lanes 16-31

    const float* arow = A + (size_t)(m0 + lm) * CC + 2 * hi; // +K offset for hi half
    const float* bcol = W + n0 + lm;

    v8f c = {};
    for (int k0 = 0; k0 < CC; k0 += 4) {
        // A: two contiguous f32 per lane (8B aligned: k0%4==0, hi adds 8B)
        v2f a;
        {
            const float2 av = *(const float2*)(arow + k0);
            a.x = av.x; a.y = av.y;
        }
        // B: one f32 per lane from rows kb and kb+1 (coalesced across lanes)
        v2f b;
        {
            int kb = k0 + 2 * hi;
            b.x = bcol[(size_t)kb * CC];
            b.y = bcol[(size_t)(kb + 1) * CC];
        }
        c = __builtin_amdgcn_wmma_f32_16x16x4_f32(
                /*neg_a=*/false, a, /*neg_b=*/false, b,
                /*c_mod=*/(short)0, c, /*reuse_a=*/false, /*reuse_b=*/false);
    }

    int col = n0 + lm;
    int rbase = m0 + 8 * hi;
#pragma unroll
    for (int j = 0; j < 8; ++j)
        out[(size_t)(rbase + j) * CC + col] = c[j];
}

// ---------------------------------------------------------------------------
// Kernel 4: backprojection. One block per (b, r, y) output row (32768 blocks).
// Stage the 2KB filtered row for the current angle in LDS; software-pipeline
// the next angle's row load; 180 cos/sin pairs precomputed per block.
// ---------------------------------------------------------------------------
__global__ void fbp_backproject(const float* __restrict__ filt,
                                float* __restrict__ out) {
    __shared__ float row[CC];
    __shared__ float cst[VV];
    __shared__ float snt[VV];

    int blk = blockIdx.x;           // (b*RR + r)*512 + y
    int y   = blk & (CC - 1);
    int br  = blk >> 9;             // b*RR + r
    int b   = br >> 5;
    int r   = br & (RR - 1);
    int tid = threadIdx.x;          // 256 threads, 2 pixels each

    if (tid < VV) {
        float ang = (float)tid * (3.14159265358979323846f / (float)VV);
        cst[tid] = cosf(ang);
        snt[tid] = sinf(ang);
    }

    const float half = (float)(CC - 1) * 0.5f;     // 255.5 (also dcenter)
    float xc0 = (float)tid - half;
    float xc1 = (float)(tid + 256) - half;
    float yc  = (float)y - half;

    size_t base = ((size_t)b * VV * RR + r) * CC;  // + v*RR*CC per angle
    const size_t vstride = (size_t)RR * CC;

    // preload angle 0
    float p0 = filt[base + tid];
    float p1 = filt[base + tid + 256];

    float acc0 = 0.0f, acc1 = 0.0f;

    for (int v = 0; v < VV; ++v) {
        __syncthreads();            // prior-iteration LDS reads complete
        row[tid]       = p0;
        row[tid + 256] = p1;
        __syncthreads();

        if (v + 1 < VV) {           // prefetch next angle's row
            size_t nb = base + (size_t)(v + 1) * vstride;
            p0 = filt[nb + tid];
            p1 = filt[nb + tid + 256];
        }

        float cv = cst[v], sv = snt[v];
        float ys = fmaf(yc, sv, half);

        float t0 = fminf(fmaxf(fmaf(xc0, cv, ys), 0.0f), (float)(CC - 1));
        int   i0 = (int)t0;
        int   j0 = min(i0 + 1, CC - 1);
        float w0 = t0 - (float)i0;
        float g00 = row[i0], g01 = row[j0];
        acc0 += g00 + w0 * (g01 - g00);

        float t1 = fminf(fmaxf(fmaf(xc1, cv, ys), 0.0f), (float)(CC - 1));
        int   i1 = (int)t1;
        int   j1 = min(i1 + 1, CC - 1);
        float w1 = t1 - (float)i1;
        float g10 = row[i1], g11 = row[j1];
        acc1 += g10 + w1 * (g11 - g10);
    }

    const float scale = 3.14159265358979323846f / (float)VV;
    size_t oidx = ((size_t)br * CC + y) * CC + tid;
    out[oidx]       = fmaxf(acc0 * scale, 0.0f);
    out[oidx + 256] = fmaxf(acc1 * scale, 0.0f);
}

// ---------------------------------------------------------------------------
extern "C" void kernel_launch(void* const* d_in, const int* in_sizes, int n_in,
                              void* d_out, int out_size, void* d_ws, size_t ws_size,
                              hipStream_t stream) {
    (void)in_sizes; (void)n_in; (void)out_size; (void)ws_size;
    const float* sino = (const float*)d_in[0];
    float* outp = (float*)d_out;

    char* ws = (char*)d_ws;
    float* h    = (float*)(ws);                       // 512 f32   (2 KB)
    float* W    = (float*)(ws + 4096);                // 512*512   (1 MB)
    float* filt = (float*)(ws + 4096 + (size_t)CC * CC * 4);  // 11520*512 (23.6 MB)

    // 1) impulse response of the ramp filter
    fbp_ramp_h<<<(CC + 255) / 256, 256, 0, stream>>>(h);

    // 2) circulant filter matrix
    fbp_circulant<<<(CC * CC) / 256, 256, 0, stream>>>(h, W);

    // 3) filtering as a WMMA f32 GEMM: (11520x512) @ (512x512)
    //    720*32 = 23040 wave-tiles, 8 waves (256 threads) per block
    fbp_filter_gemm<<<(MM / 16) * (CC / 16) / 8, 256, 0, stream>>>(sino, W, filt);

    // 4) backprojection: one block per (b, r, y)
    fbp_backproject<<<BB * RR * CC, 256, 0, stream>>>(filt, outp);
}